// DenseCapsule_24653112279600
// MI455X (gfx1250) — compile-verified
//
#include <hip/hip_runtime.h>

typedef float v2f __attribute__((ext_vector_type(2)));
typedef float v4f __attribute__((ext_vector_type(4)));
typedef float v8f __attribute__((ext_vector_type(8)));

#define B_TOT   128
#define IN_CAPS 4608
#define O_CAPS  10
#define DOUT    16
#define BG      8                       // batch groups of 16
#define NS      36                      // i-slices
#define I_PER_WG   (IN_CAPS / NS)       // 128
#define WAVES      4
#define I_PER_WAVE (I_PER_WG / WAVES)   // 32
#define BO      (B_TOT * O_CAPS)        // 1280

__device__ __forceinline__ v8f wmma_f32(v2f a, v2f b, v8f c) {
  // D = A(16x4,f32) * B(4x16,f32) + C(16x16,f32)
  return __builtin_amdgcn_wmma_f32_16x16x4_f32(false, a, false, b, (short)0, c,
                                               false, false);
}

__global__ void caps_init(float* __restrict__ vbuf, float* __restrict__ out) {
  int t = blockIdx.x * blockDim.x + threadIdx.x;
  if (t < BO * DOUT) { vbuf[t] = 0.f; out[t] = 0.f; }
}

// One WG = (batch-group bg of 16 batches) x (i-slice is of 128 i's), 4 waves.
// Per wave, per i: xh[g=16][d=16] via 2x V_WMMA_F32_16X16X4_F32 (K=8),
// logits = v . xh (one cross-half shuffle), softmax over o, accumulate c*xh.
__global__ __launch_bounds__(128) void caps_pass(
    const float* __restrict__ x, const float* __restrict__ w,
    const float* __restrict__ vbuf, float* __restrict__ part)
{
  __shared__ float sacc[WAVES * 16 * O_CAPS * DOUT];   // 40 KB

  const int tid  = threadIdx.x;
  const int wave = tid >> 5;
  const int lane = tid & 31;
  const int g    = lane & 15;          // batch-in-group (N col) == d row for A
  const int half = lane >> 4;          // 0: d 0..7 / k 0..1,  1: d 8..15 / k 2..3
  const int bg   = blockIdx.x / NS;
  const int is   = blockIdx.x % NS;
  const int b    = bg * 16 + g;        // batch this lane owns
  const int i0   = is * I_PER_WG + wave * I_PER_WAVE;

  // routing vector v[b][o][d], this lane needs d = 8*half + j
  float vr[O_CAPS][8];
  {
    const float* vb = vbuf + (size_t)b * O_CAPS * DOUT + 8 * half;
#pragma unroll
    for (int o = 0; o < O_CAPS; ++o) {
      v4f p0 = *(const v4f*)(vb + o * DOUT);
      v4f p1 = *(const v4f*)(vb + o * DOUT + 4);
      vr[o][0] = p0.x; vr[o][1] = p0.y; vr[o][2] = p0.z; vr[o][3] = p0.w;
      vr[o][4] = p1.x; vr[o][5] = p1.y; vr[o][6] = p1.z; vr[o][7] = p1.w;
    }
  }

  float sa[O_CAPS][8];
#pragma unroll
  for (int o = 0; o < O_CAPS; ++o)
#pragma unroll
    for (int j = 0; j < 8; ++j) sa[o][j] = 0.f;

  const size_t OSTR  = (size_t)IN_CAPS * 128;                 // weight o-stride
  const float* wbase = w + (size_t)g * 8 + 2 * half;          // + i*128 + o*OSTR
  const float* xbase = x + (size_t)b * IN_CAPS * 8 + 2 * half;

  for (int ii = 0; ii < I_PER_WAVE; ++ii) {
    const int i = i0 + ii;
    // B operand: X^T (8 x 16), lane supplies X[g][k], k = 2*half (+1), +4
    const float* xp = xbase + (size_t)i * 8;
    v2f bx0 = *(const v2f*)(xp);
    v2f bx1 = *(const v2f*)(xp + 4);

    const float* wp = wbase + (size_t)i * 128;
    float lg[O_CAPS];

    // ---- pass 1 over o: logits = v[o] . xh[g] ----
#pragma unroll
    for (int o = 0; o < O_CAPS; ++o) {
      const float* wo = wp + (size_t)o * OSTR;
      v2f a0 = *(const v2f*)(wo);        // A row m = d = lane&15, k = 2*half(+1)
      v2f a1 = *(const v2f*)(wo + 4);
      v8f dacc = {0.f, 0.f, 0.f, 0.f, 0.f, 0.f, 0.f, 0.f};
      dacc = wmma_f32(a0, bx0, dacc);
      dacc = wmma_f32(a1, bx1, dacc);
      float p = 0.f;
#pragma unroll
      for (int j = 0; j < 8; ++j) p += vr[o][j] * dacc[j];
      p += __shfl_xor(p, 16, 32);        // combine the two d-halves
      lg[o] = p;
    }

    // ---- softmax over the 10 output caps ----
    float m = lg[0];
#pragma unroll
    for (int o = 1; o < O_CAPS; ++o) m = fmaxf(m, lg[o]);
    float sum = 0.f;
#pragma unroll
    for (int o = 0; o < O_CAPS; ++o) { lg[o] = __expf(lg[o] - m); sum += lg[o]; }
    const float inv = 1.f / sum;

    // ---- pass 2 over o: s[o][d] += c * xh (weights re-hit L0) ----
#pragma unroll
    for (int o = 0; o < O_CAPS; ++o) {
      const float* wo = wp + (size_t)o * OSTR;
      v2f a0 = *(const v2f*)(wo);
      v2f a1 = *(const v2f*)(wo + 4);
      v8f dacc = {0.f, 0.f, 0.f, 0.f, 0.f, 0.f, 0.f, 0.f};
      dacc = wmma_f32(a0, bx0, dacc);
      dacc = wmma_f32(a1, bx1, dacc);
      const float c = lg[o] * inv;
#pragma unroll
      for (int j = 0; j < 8; ++j) sa[o][j] += c * dacc[j];
    }
  }

  // per-wave partials into LDS: [wave][g][o][d]
#pragma unroll
  for (int o = 0; o < O_CAPS; ++o) {
    float* p = &sacc[(((wave * 16 + g) * O_CAPS) + o) * DOUT + 8 * half];
    *(v4f*)(p)     = (v4f){sa[o][0], sa[o][1], sa[o][2], sa[o][3]};
    *(v4f*)(p + 4) = (v4f){sa[o][4], sa[o][5], sa[o][6], sa[o][7]};
  }
  __syncthreads();

  // fixed-order sum of the 4 wave slices -> deterministic; plain stores
  for (int e = tid; e < 16 * O_CAPS * DOUT; e += 128) {
    float s = sacc[e] + sacc[e + 2560] + sacc[e + 5120] + sacc[e + 7680];
    int gg = e / (O_CAPS * DOUT);
    int od = e % (O_CAPS * DOUT);
    part[((size_t)is * B_TOT + bg * 16 + gg) * (O_CAPS * DOUT) + od] = s;
  }
}

// One thread per (b,o): reduce NS partials, squash, update v and output.
__global__ void caps_fin(const float* __restrict__ part, float* __restrict__ vbuf,
                         float* __restrict__ out, float kcoef, int addv)
{
  int idx = blockIdx.x * blockDim.x + threadIdx.x;
  if (idx >= BO) return;
  float s[DOUT];
#pragma unroll
  for (int d = 0; d < DOUT; ++d) s[d] = 0.f;
  for (int is = 0; is < NS; ++is) {
    const float* p = part + ((size_t)is * BO + idx) * DOUT;
#pragma unroll
    for (int d = 0; d < DOUT; ++d) s[d] += p[d];
  }
  float n2 = 0.f;
#pragma unroll
  for (int d = 0; d < DOUT; ++d) n2 += s[d] * s[d];
  float n = sqrtf(n2);
  float scale = n2 / ((1.f + n2) * (n + 1e-8f));
#pragma unroll
  for (int d = 0; d < DOUT; ++d) {
    float o = scale * s[d];
    out[idx * DOUT + d] += kcoef * o;
    if (addv) vbuf[idx * DOUT + d] += o;
  }
}

extern "C" void kernel_launch(void* const* d_in, const int* in_sizes, int n_in,
                              void* d_out, int out_size, void* d_ws, size_t ws_size,
                              hipStream_t stream) {
  const float* x = (const float*)d_in[0];     // [128, 4608, 8]
  const float* w = (const float*)d_in[1];     // [10, 4608, 16, 8]
  float* out  = (float*)d_out;                // [128, 10, 16]
  float* vbuf = (float*)d_ws;                 // [128,10,16]
  float* part = vbuf + BO * DOUT;             // [NS,128,10,16]

  caps_init<<<(BO * DOUT + 255) / 256, 256, 0, stream>>>(vbuf, out);

  const float K[3] = {0.3f, 0.3f, 0.4f};
  for (int it = 0; it < 3; ++it) {
    caps_pass<<<BG * NS, 128, 0, stream>>>(x, w, vbuf, part);
    caps_fin<<<(BO + 127) / 128, 128, 0, stream>>>(part, vbuf, out, K[it],
                                                   it < 2 ? 1 : 0);
  }
}